// BuiltCCNOT_31662498906411
// MI455X (gfx1250) — compile-verified
//
#include <hip/hip_runtime.h>
#include <stdint.h>

// CCNOT (c1=0,c2=1,t=2, 12 qubits) applied as state @ M where M is a
// permutation matrix:  out[b, j] = state[b, j ^ 0x200] iff bits 11&10 of j
// are set, else state[b, j].  Pure bandwidth problem: 32 MiB in + 32 MiB out
// => ~2.9 us at 23.3 TB/s.  Implemented via the CDNA5 async data-mover path
// (global_load_async_to_lds_b128 / global_store_async_from_lds_b128,
// ASYNCcnt-tracked) so data never transits VGPRs.
//
// float4-granularity mapping: j4 = j >> 2.  bit11(j)->bit9(j4),
// bit10(j)->bit8(j4), bit9(j)->bit7(j4);  swap = 0x200/4 = 1<<7.

#define ROW_F4 1024u        // float4 per 4096-float row
#define THREADS 256u
#define F4_PER_THREAD 4u    // 256 threads * 4 = 1024 float4 = one row / block

__global__ __launch_bounds__(256)
void ccnot_permute_async(const float4* __restrict__ in, float4* __restrict__ out) {
  __shared__ float4 stage[ROW_F4];   // 16 KB staging for one row

  const uint32_t tid = threadIdx.x;
  const uint32_t row = blockIdx.x;
  const uint64_t row_base = (uint64_t)row * ROW_F4;      // in float4 units
  const uint32_t lds_base = (uint32_t)(uintptr_t)(&stage[0]); // LDS byte offset

  // Phase 1: async gather: LDS slot j4 gets this row's input float4 j4, bit 7 flipped if bits 9&8 set
#pragma unroll
  for (uint32_t i = 0; i < F4_PER_THREAD; ++i) {
    const uint32_t j4   = i * THREADS + tid;                    // 0..1023
    const uint32_t swap = ((j4 >> 9) & (j4 >> 8) & 1u) << 7;    // controls -> flip t
    const uint32_t src4 = j4 ^ swap;
    const uint64_t src_addr = (uint64_t)(uintptr_t)in + ((row_base + src4) << 4);
    const uint32_t lds_addr = lds_base + (j4 << 4);
    asm volatile("global_load_async_to_lds_b128 %0, %1, off"
                 :: "v"(lds_addr), "v"(src_addr) : "memory");
  }

  // Wave-local: all of this wave's async loads have landed in LDS.
  // Each wave only reads back LDS slots it wrote itself -> no barrier needed.
  asm volatile("s_wait_asynccnt 0x0" ::: "memory");

  // Phase 2: async store: LDS slot j4 goes to this row's output float4 j4
#pragma unroll
  for (uint32_t i = 0; i < F4_PER_THREAD; ++i) {
    const uint32_t j4 = i * THREADS + tid;
    const uint64_t dst_addr = (uint64_t)(uintptr_t)out + ((row_base + j4) << 4);
    const uint32_t lds_addr = lds_base + (j4 << 4);
    asm volatile("global_store_async_from_lds_b128 %0, %1, off"
                 :: "v"(dst_addr), "v"(lds_addr) : "memory");
  }

  // Ensure stores have left before wave teardown (S_ENDPGM also wait-idles).
  asm volatile("s_wait_asynccnt 0x0" ::: "memory");
}

extern "C" void kernel_launch(void* const* d_in, const int* in_sizes, int n_in,
                              void* d_out, int out_size, void* d_ws, size_t ws_size,
                              hipStream_t stream) {
  (void)n_in; (void)d_ws; (void)ws_size; (void)out_size;
  const float4* state = (const float4*)d_in[0];   // [BATCH, 4096] fp32
  // d_in[1] (M) is deliberately unused: it is a fixed permutation.
  float4* out = (float4*)d_out;

  const int rows = in_sizes[0] / 4096;            // BATCH (2048)
  dim3 grid((unsigned)rows), block(THREADS);
  ccnot_permute_async<<<grid, block, 0, stream>>>(state, out);
}